// DistGCN_39908836114884
// MI455X (gfx1250) — compile-verified
//
#include <hip/hip_runtime.h>

// ---------------------------------------------------------------------------
// CDNA5 (gfx1250) GCN forward: fp32 WMMA GEMMs + L2-resident atomic scatter.
// ---------------------------------------------------------------------------

typedef __attribute__((ext_vector_type(2))) float v2f;
typedef __attribute__((ext_vector_type(8))) float v8f;

// ---- in-degree: deg[dst[e]] += 1 ------------------------------------------
__global__ void gcn_degree_kernel(const int* __restrict__ dst,
                                  float* __restrict__ deg, int E) {
  int t = blockIdx.x * blockDim.x + threadIdx.x;
  if (t < E) unsafeAtomicAdd(&deg[dst[t]], 1.0f);
}

// ---- dense GEMM Y[N,M] = X[N,K] @ W[K,M] via V_WMMA_F32_16X16X4_F32 -------
// One wave owns one 16-row block and ALL M = TM*16 columns: the A fragment is
// loaded once per k-step and reused across TM independent accumulators
// (back-to-back WMMAs on different D registers -> good XDL pipelining).
//
// A (16x4 f32): lanes 0-15 hold rows with K={k,k+1} in VGPR0/1; lanes 16-31
// the same rows with K={k+2,k+3}. B (4x16) mirrors per column. C/D: VGPR r
// holds (M = r + 8*(lane>=16), N = lane&15).  Requires N%16==0, K%4==0.
template <int K, int TM>
__global__ void gcn_gemm_wmma_f32(const float* __restrict__ X,
                                  const float* __restrict__ W,
                                  float* __restrict__ Y, int N) {
  const int M = TM * 16;
  const int wave = (int)((blockIdx.x * (size_t)blockDim.x + threadIdx.x) >> 5);
  if (wave >= (N >> 4)) return;          // wave-uniform: EXEC stays all-1s

  const int lane = threadIdx.x & 31;
  const int half = lane >> 4;            // 0: K pair {0,1}; 1: K pair {2,3}
  const int l16  = lane & 15;
  const float* __restrict__ xrow = X + (size_t)((wave << 4) + l16) * K;

  v8f acc[TM];
#pragma unroll
  for (int j = 0; j < TM; ++j) acc[j] = (v8f){};

#pragma unroll 2
  for (int k = 0; k < K; k += 4) {
    const int kk = k + (half << 1);
    const v2f a = *(const v2f*)(xrow + kk);           // contiguous, 8B aligned
#pragma unroll
    for (int j = 0; j < TM; ++j) {
      v2f b;
      b.x = W[kk * M + (j << 4) + l16];               // coalesced across lanes
      b.y = W[(kk + 1) * M + (j << 4) + l16];
      acc[j] = __builtin_amdgcn_wmma_f32_16x16x4_f32(
          /*neg_a=*/false, a, /*neg_b=*/false, b,
          /*c_mod=*/(short)0, acc[j], /*reuse_a=*/false, /*reuse_b=*/false);
    }
  }

  const int rbase = (wave << 4) + (half << 3);
#pragma unroll
  for (int j = 0; j < TM; ++j)
#pragma unroll
    for (int r = 0; r < 8; ++r)
      Y[(size_t)(rbase + r) * M + (j << 4) + l16] = acc[j][r];
}

// ---- edge scatter-add: agg[dst[e],:] += h[src[e],:] -----------------------
// width = 4 << SHIFT floats; (1<<SHIFT) threads per edge, float4 per thread.
template <int SHIFT>
__global__ void gcn_scatter_add(const float* __restrict__ h,
                                const int* __restrict__ src,
                                const int* __restrict__ dst,
                                float* __restrict__ agg, int E) {
  const int width = 4 << SHIFT;
  long long t = (long long)blockIdx.x * blockDim.x + threadIdx.x;
  int e = (int)(t >> SHIFT);
  if (e >= E) return;
  int c = ((int)t & ((1 << SHIFT) - 1)) << 2;
  int s = src[e], d = dst[e];
  float4 v = *(const float4*)(h + (size_t)s * width + c);   // L2-resident gather
  float* ap = agg + (size_t)d * width + c;
  unsafeAtomicAdd(ap + 0, v.x);    // global_atomic_add_f32, no return
  unsafeAtomicAdd(ap + 1, v.y);
  unsafeAtomicAdd(ap + 2, v.z);
  unsafeAtomicAdd(ap + 3, v.w);
}

// ---- finalize: out = agg / max(deg,1) + bias, optional relu ---------------
template <int LOGW>
__global__ void gcn_finalize(const float* __restrict__ agg,
                             const float* __restrict__ deg,
                             const float* __restrict__ bias,
                             float* __restrict__ out, int N, int doRelu) {
  const int width = 1 << LOGW;
  long long t = (long long)blockIdx.x * blockDim.x + threadIdx.x;
  if (t >= ((long long)N << LOGW)) return;
  int n = (int)(t >> LOGW);
  int c = (int)t & (width - 1);
  float d = deg[n];
  d = d < 1.0f ? 1.0f : d;
  float r = agg[t] / d + bias[c];
  if (doRelu) r = r > 0.0f ? r : 0.0f;
  out[t] = r;
}

// ---------------------------------------------------------------------------
extern "C" void kernel_launch(void* const* d_in, const int* in_sizes, int n_in,
                              void* d_out, int out_size, void* d_ws, size_t ws_size,
                              hipStream_t stream) {
  const float* feat = (const float*)d_in[0];   // [N,256]
  const float* W0   = (const float*)d_in[1];   // [256,64]
  const float* b0   = (const float*)d_in[2];   // [64]
  const float* W1   = (const float*)d_in[3];   // [64,64]
  const float* b1   = (const float*)d_in[4];   // [64]
  const float* W2   = (const float*)d_in[5];   // [64,32]
  const float* b2   = (const float*)d_in[6];   // [32]
  const int*   src  = (const int*)d_in[7];     // [E]
  const int*   dst  = (const int*)d_in[8];     // [E]
  float* out = (float*)d_out;                  // [N,32]

  const int N = in_sizes[0] / 256;             // 100000 (divisible by 16)
  const int E = in_sizes[7];                   // 1600000

  // Workspace layout: bufA[N*64] | bufB[N*64] | deg[N]  (~51.6 MB)
  float* bufA = (float*)d_ws;
  float* bufB = bufA + (size_t)N * 64;
  float* deg  = bufB + (size_t)N * 64;
  const size_t sz64 = (size_t)N * 64 * sizeof(float);
  const size_t sz32 = (size_t)N * 32 * sizeof(float);

  const int TB = 256;
  const int gemmBlocks   = (int)((((long long)(N / 16)) * 32 + TB - 1) / TB);
  const int scat64Blocks = (int)(((long long)E * 16 + TB - 1) / TB);
  const int scat32Blocks = (int)(((long long)E * 8 + TB - 1) / TB);
  const int fin64Blocks  = (int)(((long long)N * 64 + TB - 1) / TB);
  const int fin32Blocks  = (int)(((long long)N * 32 + TB - 1) / TB);

  // ---- degree (shared by all layers; graph is fixed) ----
  hipMemsetAsync(deg, 0, (size_t)N * sizeof(float), stream);
  gcn_degree_kernel<<<(E + TB - 1) / TB, TB, 0, stream>>>(dst, deg, E);

  // ---- Layer 0: pre-transform (256 -> 64), aggregate, norm, +b0, relu ----
  gcn_gemm_wmma_f32<256, 4><<<gemmBlocks, TB, 0, stream>>>(feat, W0, bufA, N);
  hipMemsetAsync(bufB, 0, sz64, stream);
  gcn_scatter_add<4><<<scat64Blocks, TB, 0, stream>>>(bufA, src, dst, bufB, E);
  gcn_finalize<6><<<fin64Blocks, TB, 0, stream>>>(bufB, deg, b0, bufA, N, 1); // x1 -> bufA

  // ---- Layer 1: aggregate first, then matmul (64 -> 64), norm, +b1, relu ----
  hipMemsetAsync(bufB, 0, sz64, stream);
  gcn_scatter_add<4><<<scat64Blocks, TB, 0, stream>>>(bufA, src, dst, bufB, E);
  gcn_gemm_wmma_f32<64, 4><<<gemmBlocks, TB, 0, stream>>>(bufB, W1, bufA, N);
  gcn_finalize<6><<<fin64Blocks, TB, 0, stream>>>(bufA, deg, b1, bufB, N, 1); // x2 -> bufB

  // ---- Layer 2: pre-transform (64 -> 32), aggregate, norm, +b2 ----
  gcn_gemm_wmma_f32<64, 2><<<gemmBlocks, TB, 0, stream>>>(bufB, W2, bufA, N);
  hipMemsetAsync(bufB, 0, sz32, stream);
  gcn_scatter_add<3><<<scat32Blocks, TB, 0, stream>>>(bufA, src, dst, bufB, E);
  gcn_finalize<5><<<fin32Blocks, TB, 0, stream>>>(bufB, deg, b2, out, N, 0);
}